// GroupQueryAttention_25941602468243
// MI455X (gfx1250) — compile-verified
//
#include <hip/hip_runtime.h>
#include <hip/hip_bf16.h>

typedef __attribute__((ext_vector_type(2))) float v2f;
typedef __attribute__((ext_vector_type(4))) float v4f;
typedef __attribute__((ext_vector_type(8))) float v8f;
typedef __attribute__((ext_vector_type(4))) int   v4i;

// Exact-fp32 matrix op on the CDNA5 matrix pipe: D(16x16) = A(16x4) * B(4x16) + C
static __device__ __forceinline__ v8f wmma4(v2f a, v2f b, v8f c) {
    return __builtin_amdgcn_wmma_f32_16x16x4_f32(
        /*neg_a=*/false, a, /*neg_b=*/false, b,
        /*c_mod=*/(short)0, c, /*reuse_a=*/false, /*reuse_b=*/false);
}

// ---- CDNA5 async global->LDS copy (ASYNCcnt) with portable fallback -------
#if defined(__HIP_DEVICE_COMPILE__) && __has_builtin(__builtin_amdgcn_global_load_async_to_lds_b128)
#define GQA_ASYNC_LDS 1
static __device__ __forceinline__ void async_cp16(const float* g, float* l) {
    __builtin_amdgcn_global_load_async_to_lds_b128(
        (__attribute__((address_space(1))) v4i*)g,
        (__attribute__((address_space(3))) v4i*)l,
        /*imm offset=*/0, /*cpol=*/0);
}
template <int N>
static __device__ __forceinline__ void wait_async() {
#if __has_builtin(__builtin_amdgcn_s_wait_asynccnt)
    __builtin_amdgcn_s_wait_asynccnt(N);
#else
    asm volatile("s_wait_asynccnt %0" ::"i"(N) : "memory");
#endif
}
#else
#define GQA_ASYNC_LDS 0
#endif

// -------------------------------------------------------------------------
// C[M,N] = A[M,K] @ W[N,K]^T   (torch Linear convention, all fp32 row-major)
// Block tile 128x128, BK=32. 256 threads = 8 waves; wave (mw,nw) in 4x2 grid
// owns a 32x64 sub-tile = 2x4 WMMA accumulators.
// A-frag element: A[m = lane&15][k = v + 2*(lane>>4)]  -> float2 LDS load
// B-frag element: B[k = v + 2*(lane>>4)][n = lane&15] == W[n][k] -> same load
// C element:      C[m = v + 8*(lane>>4)][n = lane&15]
// Pipeline: async double-buffered LDS tiles (64 KiB) when the gfx1250
// async-to-LDS builtin is available; register-staged prefetch otherwise.
// -------------------------------------------------------------------------
static __device__ __forceinline__ void gemm_tile_compute(
    const float* __restrict__ As, const float* __restrict__ Bs,
    int mBase, int nBase, int ln, int hi, v8f acc[2][4]) {
    #pragma unroll
    for (int ks = 0; ks < 8; ++ks) {
        const int k = ks * 4;
        v2f af[2], bf[4];
        #pragma unroll
        for (int i = 0; i < 2; ++i)
            af[i] = *(const v2f*)&As[(mBase + i * 16 + ln) * 32 + k + 2 * hi];
        #pragma unroll
        for (int j = 0; j < 4; ++j)
            bf[j] = *(const v2f*)&Bs[(nBase + j * 16 + ln) * 32 + k + 2 * hi];
        #pragma unroll
        for (int i = 0; i < 2; ++i)
            #pragma unroll
            for (int j = 0; j < 4; ++j)
                acc[i][j] = wmma4(af[i], bf[j], acc[i][j]);
    }
}

__global__ __launch_bounds__(256) void gqa_gemm_nt(
    const float* __restrict__ A, const float* __restrict__ W,
    float* __restrict__ C, int M, int N, int K) {
    extern __shared__ float smem[];  // async: 2*(4096+4096) fl = 64 KiB
    (void)M;

    const int tid  = threadIdx.x;
    const int lane = tid & 31;
    const int wave = tid >> 5;
    const int hi   = lane >> 4;
    const int ln   = lane & 15;

    const int m0 = blockIdx.y * 128;
    const int n0 = blockIdx.x * 128;
    const int mBase = (wave >> 1) * 32;
    const int nBase = (wave & 1) * 64;

    const int lr = tid >> 3;        // 0..31 : row within pass
    const int lc = (tid & 7) * 4;   // 0..28 : float4 column
    const int T  = K / 32;

    v8f acc[2][4] = {};

#if GQA_ASYNC_LDS
    auto issue = [&](int t, int b) {
        const int k0 = t * 32;
        float* As = smem + b * 8192;
        float* Bs = As + 4096;
        #pragma unroll
        for (int p = 0; p < 4; ++p) {
            const int row = p * 32 + lr;
            async_cp16(&A[(size_t)(m0 + row) * K + k0 + lc], &As[row * 32 + lc]);
            async_cp16(&W[(size_t)(n0 + row) * K + k0 + lc], &Bs[row * 32 + lc]);
        }
    };
    issue(0, 0);
    for (int t = 0; t < T; ++t) {
        const int b = t & 1;
        if (t + 1 < T) {
            issue(t + 1, b ^ 1);   // prefetch into the other buffer
            wait_async<8>();       // tile t landed; tile t+1 still in flight
        } else {
            wait_async<0>();
        }
        __syncthreads();
        const float* As = smem + b * 8192;
        gemm_tile_compute(As, As + 4096, mBase, nBase, ln, hi, acc);
        __syncthreads();           // all waves done reading before overwrite
    }
#else
    float* As = smem;
    float* Bs = smem + 4096;
    v4f ra[4], rb[4];
    auto gload = [&](int t) {
        const int k0 = t * 32;
        #pragma unroll
        for (int p = 0; p < 4; ++p) {
            const int row = p * 32 + lr;
            ra[p] = *(const v4f*)&A[(size_t)(m0 + row) * K + k0 + lc];
            rb[p] = *(const v4f*)&W[(size_t)(n0 + row) * K + k0 + lc];
        }
    };
    gload(0);
    for (int t = 0; t < T; ++t) {
        __syncthreads();           // previous tile fully consumed
        #pragma unroll
        for (int p = 0; p < 4; ++p) {
            const int row = p * 32 + lr;
            *(v4f*)&As[row * 32 + lc] = ra[p];
            *(v4f*)&Bs[row * 32 + lc] = rb[p];
        }
        if (t + 1 < T) gload(t + 1);  // overlap next tile's loads with compute
        __syncthreads();
        gemm_tile_compute(As, Bs, mBase, nBase, ln, hi, acc);
    }
#endif

    #pragma unroll
    for (int i = 0; i < 2; ++i)
        #pragma unroll
        for (int j = 0; j < 4; ++j)
            #pragma unroll
            for (int v = 0; v < 8; ++v)
                C[(size_t)(m0 + mBase + i * 16 + v + 8 * hi) * N +
                  (n0 + nBase + j * 16 + ln)] = acc[i][j][v];
}

// -------------------------------------------------------------------------
// Flash-style GQA attention core.
// Grid: (16 q-blocks of 128 rows, 32 heads). 256 threads = 8 waves, each wave
// owns 16 query rows. Q kept in registers in A-fragment layout (pre-scaled by
// 1/sqrt(128)). 32-key K/V tiles streamed row-major through LDS with
// register-prefetch double buffering. Online softmax per 16-row strip; P is
// staged through wave-private LDS to convert C-layout -> A-layout.
// -------------------------------------------------------------------------
__global__ __launch_bounds__(256) void gqa_attention(
    const float* __restrict__ Q,   // [2048, 4096]
    const float* __restrict__ Kp,  // [2048, 1024]
    const float* __restrict__ Vp,  // [2048, 1024]
    float* __restrict__ O) {       // [2048, 4096]
    extern __shared__ float smem[];
    float* Ks = smem;              // [32][128]
    float* Vs = smem + 4096;       // [32][128]
    float* Ps = smem + 8192;       // 8 waves x [16][32]

    const int tid  = threadIdx.x;
    const int lane = tid & 31;
    const int wave = tid >> 5;
    const int hi   = lane >> 4;
    const int ln   = lane & 15;

    const int qb  = blockIdx.x;
    const int h   = blockIdx.y;
    const int kvh = h >> 2;             // GQA: 4 Q heads share a KV head
    const int qRow0 = qb * 128 + wave * 16;
    float* Pw = &Ps[wave * 16 * 32];

    // This wave's 16x128 Q strip -> A-fragment registers, pre-scaled.
    const float sm = 0.08838834764831845f;  // 1/sqrt(128)
    v2f qf[32];
    {
        const float* qp = Q + (size_t)(qRow0 + ln) * 4096 + h * 128 + 2 * hi;
        #pragma unroll
        for (int kk = 0; kk < 32; ++kk) {
            v2f t = *(const v2f*)(qp + kk * 4);
            qf[kk] = t * sm;
        }
    }

    v8f oacc[8] = {};
    float mrow[8], lrow[8];
    #pragma unroll
    for (int v = 0; v < 8; ++v) { mrow[v] = -INFINITY; lrow[v] = 0.0f; }

    const int lr = tid >> 5;         // 0..7 : row within pass
    const int lc = (tid & 31) * 4;   // 0..124 : float4 column

    v4f rk[4], rv[4];
    auto gload = [&](int kb) {
        #pragma unroll
        for (int p = 0; p < 4; ++p) {
            const int r = p * 8 + lr;
            const size_t g = (size_t)(kb * 32 + r) * 1024 + kvh * 128 + lc;
            rk[p] = *(const v4f*)&Kp[g];
            rv[p] = *(const v4f*)&Vp[g];
        }
    };
    gload(0);

    for (int kb = 0; kb < 64; ++kb) {
        __syncthreads();             // previous K/V tile fully consumed
        #pragma unroll
        for (int p = 0; p < 4; ++p) {
            const int r = p * 8 + lr;
            *(v4f*)&Ks[r * 128 + lc] = rk[p];
            *(v4f*)&Vs[r * 128 + lc] = rv[p];
        }
        if (kb + 1 < 64) gload(kb + 1);  // overlap next tile with compute
        __syncthreads();

        // S(16x32) = Qstrip(16x128) @ Ktile(32x128)^T  — 64 WMMAs
        v8f sacc[2] = {};
        #pragma unroll
        for (int kk = 0; kk < 32; ++kk) {
            const v2f a = qf[kk];
            #pragma unroll
            for (int j = 0; j < 2; ++j) {
                const v2f b = *(const v2f*)&Ks[(j * 16 + ln) * 128 + kk * 4 + 2 * hi];
                sacc[j] = wmma4(a, b, sacc[j]);
            }
        }

        // Online softmax update per C-layout row (row = v + 8*hi).
        #pragma unroll
        for (int v = 0; v < 8; ++v) {
            const float s0 = sacc[0][v];
            const float s1 = sacc[1][v];
            float m = fmaxf(s0, s1);
            m = fmaxf(m, __shfl_xor(m, 1, 32));
            m = fmaxf(m, __shfl_xor(m, 2, 32));
            m = fmaxf(m, __shfl_xor(m, 4, 32));
            m = fmaxf(m, __shfl_xor(m, 8, 32));
            const float newm  = fmaxf(mrow[v], m);
            const float scale = __expf(mrow[v] - newm);
            const float p0 = __expf(s0 - newm);
            const float p1 = __expf(s1 - newm);
            float ps = p0 + p1;
            ps += __shfl_xor(ps, 1, 32);
            ps += __shfl_xor(ps, 2, 32);
            ps += __shfl_xor(ps, 4, 32);
            ps += __shfl_xor(ps, 8, 32);
            lrow[v] = lrow[v] * scale + ps;
            mrow[v] = newm;
            #pragma unroll
            for (int j = 0; j < 8; ++j) oacc[j][v] *= scale;
            const int r = v + 8 * hi;     // C-layout row -> LDS row
            Pw[r * 32 + ln]      = p0;
            Pw[r * 32 + 16 + ln] = p1;
        }
        // Wave-private staging: same-wave LDS ops are in order, no barrier.

        // O(16x128) += P(16x32) @ Vtile(32x128)  — 64 WMMAs
        #pragma unroll
        for (int ks = 0; ks < 8; ++ks) {
            const v2f aP = *(const v2f*)&Pw[ln * 32 + ks * 4 + 2 * hi];
            const int k = ks * 4 + 2 * hi;
            #pragma unroll
            for (int j = 0; j < 8; ++j) {
                const int n = j * 16 + ln;
                v2f bV;
                bV.x = Vs[k * 128 + n];
                bV.y = Vs[(k + 1) * 128 + n];
                oacc[j] = wmma4(aP, bV, oacc[j]);
            }
        }
    }

    // Normalize and write [S, nH*hd] with head h at columns h*128..h*128+127.
    #pragma unroll
    for (int v = 0; v < 8; ++v) {
        const float inv = 1.0f / lrow[v];
        const size_t row = (size_t)(qRow0 + v + 8 * hi);
        #pragma unroll
        for (int j = 0; j < 8; ++j)
            O[row * 4096 + h * 128 + j * 16 + ln] = oacc[j][v] * inv;
    }
}

extern "C" void kernel_launch(void* const* d_in, const int* in_sizes, int n_in,
                              void* d_out, int out_size, void* d_ws, size_t ws_size,
                              hipStream_t stream) {
    (void)in_sizes; (void)n_in; (void)out_size; (void)ws_size;
    const float* X  = (const float*)d_in[0];  // [2048, 4096]
    const float* Wq = (const float*)d_in[1];  // [4096, 4096]
    const float* Wk = (const float*)d_in[2];  // [1024, 4096]
    const float* Wv = (const float*)d_in[3];  // [1024, 4096]
    const float* Wo = (const float*)d_in[4];  // [4096, 4096]
    float* out = (float*)d_out;               // [2048, 4096]

    float* ws = (float*)d_ws;
    float* Qp = ws;                           // 2048*4096 = 8388608
    float* Kp = ws + 8388608;                 // 2048*1024 = 2097152
    float* Vp = ws + 10485760;                // 2048*1024
    float* AO = ws + 12582912;                // 2048*4096
    // total workspace: 20971520 floats = 80 MiB

    const size_t gemmSmem = 65536;  // double-buffered 128x32 A/B tiles
    const size_t attnSmem = 49152;  // K + V + P staging

    // Projections: x @ W^T
    gqa_gemm_nt<<<dim3(32, 16), 256, gemmSmem, stream>>>(X, Wq, Qp, 2048, 4096, 4096);
    gqa_gemm_nt<<<dim3(8, 16),  256, gemmSmem, stream>>>(X, Wk, Kp, 2048, 1024, 4096);
    gqa_gemm_nt<<<dim3(8, 16),  256, gemmSmem, stream>>>(X, Wv, Vp, 2048, 1024, 4096);
    // Attention core
    gqa_attention<<<dim3(16, 32), 256, attnSmem, stream>>>(Qp, Kp, Vp, AO);
    // Output projection
    gqa_gemm_nt<<<dim3(32, 16), 256, gemmSmem, stream>>>(AO, Wo, out, 2048, 4096, 4096);
}